// OverlapTimeWindowLayer_309237646059
// MI455X (gfx1250) — compile-verified
//
#include <hip/hip_runtime.h>

#define AS1 __attribute__((address_space(1)))
#define AS3 __attribute__((address_space(3)))

typedef int v2i __attribute__((vector_size(8)));
typedef int v4i __attribute__((vector_size(16)));

constexpr int B = 32;
constexpr int T = 4096;
constexpr int C = 30;
constexpr int W = 19;
constexpr int PAD = W / 2;              // 9
constexpr int TTILE = 128;              // time steps per block
constexpr int TILES_PER_B = T / TTILE;  // 32
constexpr int ROWS = TTILE + 2 * PAD;   // 146 staged source rows
constexpr int LDS_FLOATS = ROWS * C;    // 4380 floats = 17520 B
constexpr int LDS_PAIRS = LDS_FLOATS / 2;  // 2190 8-byte pairs
constexpr int WC = W * C;               // 570 floats per output row
constexpr int ROW_BYTES = WC * 4;       // 2280
constexpr int SRC_ROW_BYTES = C * 4;    // 120
constexpr int NCHUNK = TTILE * WC / 4;  // 18240 float4 chunks per block
constexpr int NTHREADS = 256;

__global__ __launch_bounds__(NTHREADS) void overlap_window_kernel(
    const float* __restrict__ x, float* __restrict__ out) {
#if defined(__HIP_DEVICE_COMPILE__)
  __shared__ __align__(16) float s[LDS_FLOATS];

  const int blk  = blockIdx.x;
  const int b    = blk / TILES_PER_B;
  const int tile = blk - b * TILES_PER_B;
  const int t0   = tile * TTILE;
  const int r0c  = (t0 - PAD) * C;  // first staged source element (even, may be < 0)
  const int tid  = threadIdx.x;

  const float* xb = x + (size_t)b * T * C;

  // ---- Stage (TTILE + 2*PAD) x C source floats into LDS, 8B granularity --
  // Pair p covers source elements {2p + r0c, 2p + r0c + 1}; validity never
  // splits a pair (r0c and T*C are both even). Out-of-range pairs are the
  // zero padding at the t=0 / t=T edges.
  for (int p = tid; p < LDS_PAIRS; p += NTHREADS) {
    const int g = 2 * p + r0c;
    if (g < 0 || g >= T * C) {
      *(float2*)&s[2 * p] = make_float2(0.0f, 0.0f);  // DS store; barrier waits DScnt
    } else {
#if __has_builtin(__builtin_amdgcn_global_load_async_to_lds_b64)
      __builtin_amdgcn_global_load_async_to_lds_b64(
          (AS1 v2i*)(AS1 void*)(void*)const_cast<float*>(xb + g),
          (AS3 v2i*)(AS3 void*)(void*)&s[2 * p],
          /*imm_offset=*/0, /*cpol=*/0);
#else
      s[2 * p]     = xb[g];
      s[2 * p + 1] = xb[g + 1];
#endif
    }
  }
  asm volatile("s_wait_asynccnt 0" ::: "memory");
  __syncthreads();

  // ---- Emit output: per t, a contiguous 2280B LDS slice -> global --------
  // out[t, w*30+c] == lds[(t-t0)*30 + (w*30+c)]  (overlapped sliding copy).
  // Chunk o4 covers output bytes [o4*16, o4*16+16). LDS source byte
  // = o4*16 - t_local*2160 (2160 = 135*16) -> 16-aligned on both sides, so
  // the fast path is one async LDS->global b128 DMA per chunk.
  float* ob = out + (size_t)(b * T + t0) * WC;
  char* sbytes = (char*)s;

  for (int o4 = tid; o4 < NCHUNK; o4 += NTHREADS) {
    const int byte = o4 * 16;
    const int tl   = byte / ROW_BYTES;           // const-divisor -> mul/shift
    const int r    = byte - tl * ROW_BYTES;
    char* gdst = (char*)ob + byte;
    if (r + 16 <= ROW_BYTES) {
      char* lsrc = sbytes + tl * SRC_ROW_BYTES + r;   // 16-aligned (proven)
#if __has_builtin(__builtin_amdgcn_global_store_async_from_lds_b128)
      __builtin_amdgcn_global_store_async_from_lds_b128(
          (AS1 v4i*)(AS1 void*)(void*)gdst,
          (AS3 v4i*)(AS3 void*)(void*)lsrc,
          /*imm_offset=*/0, /*cpol=*/0);
#else
      *(float4*)gdst = *(const float4*)lsrc;
#endif
    } else {
      // Rare (64/18240): chunk straddles an output-row boundary; the LDS
      // source jumps backward by 18 rows mid-chunk. Gather 4 floats.
      float e0, e1, e2, e3;
      {
        int gb = byte;      int tk = gb / ROW_BYTES; int rk = gb - tk * ROW_BYTES;
        e0 = *(const float*)(sbytes + tk * SRC_ROW_BYTES + rk);
      }
      {
        int gb = byte + 4;  int tk = gb / ROW_BYTES; int rk = gb - tk * ROW_BYTES;
        e1 = *(const float*)(sbytes + tk * SRC_ROW_BYTES + rk);
      }
      {
        int gb = byte + 8;  int tk = gb / ROW_BYTES; int rk = gb - tk * ROW_BYTES;
        e2 = *(const float*)(sbytes + tk * SRC_ROW_BYTES + rk);
      }
      {
        int gb = byte + 12; int tk = gb / ROW_BYTES; int rk = gb - tk * ROW_BYTES;
        e3 = *(const float*)(sbytes + tk * SRC_ROW_BYTES + rk);
      }
      *(float4*)gdst = make_float4(e0, e1, e2, e3);
    }
  }
  // Async stores drain before wave exit via the implicit S_WAIT_IDLE at
  // S_ENDPGM; LDS stays allocated until the wave terminates.
#else
  (void)x; (void)out;
#endif
}

extern "C" void kernel_launch(void* const* d_in, const int* in_sizes, int n_in,
                              void* d_out, int out_size, void* d_ws, size_t ws_size,
                              hipStream_t stream) {
  (void)in_sizes; (void)n_in; (void)out_size; (void)d_ws; (void)ws_size;
  const float* x = (const float*)d_in[0];
  float* out = (float*)d_out;
  const dim3 grid(B * TILES_PER_B);   // 1024 blocks
  const dim3 block(NTHREADS);         // 8 wave32s
  overlap_window_kernel<<<grid, block, 0, stream>>>(x, out);
}